// RNNLM_18700287607668
// MI455X (gfx1250) — compile-verified
//
#include <hip/hip_runtime.h>

typedef __attribute__((ext_vector_type(2))) float v2f;
typedef __attribute__((ext_vector_type(4))) float v4f;
typedef __attribute__((ext_vector_type(8))) float v8f;

#define SEQ_LEN 8192
#define VOCAB   128000
#define EMB     10
#define HID     10
#define CW      20                    // combined width (EMB+HID)
#define NSPLIT  8                     // vocab splits per time-tile pair
#define VTILES  (VOCAB / 16)          // 8000
#define TTILES  (SEQ_LEN / 16)        // 512
#define TILES_PER_SPLIT (VTILES / NSPLIT)  // 1000

// ---------------------------------------------------------------------------
// Kernel 1: gather embeddings into Combined[:,0:10] and precompute
// c_t = W_i2h[:,0:10] @ emb_t + b_i2h  (removes the embedding contribution
// from the serial recurrence's critical path).
// ---------------------------------------------------------------------------
__global__ void k_embed(const int* __restrict__ words,
                        const float* __restrict__ emb,
                        const float* __restrict__ W_i2h,
                        const float* __restrict__ b_i2h,
                        float* __restrict__ Cmat,
                        float* __restrict__ cvec) {
  int t = blockIdx.x * blockDim.x + threadIdx.x;
  if (t >= SEQ_LEN) return;
  int w = words[t];
  const float* ep = emb + (long)w * EMB;
  float e[EMB];
#pragma unroll
  for (int j = 0; j < EMB; j++) { e[j] = ep[j]; Cmat[t * CW + j] = e[j]; }
#pragma unroll
  for (int i = 0; i < HID; i++) {
    float acc = b_i2h[i];
#pragma unroll
    for (int j = 0; j < EMB; j++) acc += W_i2h[i * CW + j] * e[j];
    cvec[t * HID + i] = acc;
  }
}

// ---------------------------------------------------------------------------
// Kernel 2: serial linear recurrence h_{t+1} = A h_t + c_t, one wave.
// Lane i (<10) owns h[i] and row i of A = W_i2h[:,10:20]; per step the chain
// is 10 dependent FMAs + shuffles. Writes h_t into Combined[:,10:20].
// ---------------------------------------------------------------------------
__global__ void k_recur(const float* __restrict__ W_i2h,
                        const float* __restrict__ cvec,
                        float* __restrict__ Cmat) {
  int lane = threadIdx.x;  // 0..31, wave32
  float a[HID];
#pragma unroll
  for (int j = 0; j < HID; j++)
    a[j] = (lane < HID) ? W_i2h[lane * CW + EMB + j] : 0.0f;

  float h = 0.0f;
  for (int t = 0; t < SEQ_LEN; t++) {
    if (lane < HID) Cmat[t * CW + EMB + lane] = h;
    float c = 0.0f;
    if (lane < HID) c = cvec[t * HID + lane];
    float acc = c;
#pragma unroll
    for (int j = 0; j < HID; j++) acc += a[j] * __shfl(h, j, 32);
    h = acc;
  }
}

// ---------------------------------------------------------------------------
// Kernel 3: fused GEMM + argmax, 2 time-tiles per wave.
// D = W_io_tile(16v x 20k) x Combined^T(20k x 16t) via 5x WMMA f32 16x16x4
// per time tile; A-operand and bias are vocab-only so one set of 7 loads
// feeds 10 WMMAs. Accumulators are bias-initialized (bias is free).
// Per-lane running (max, argidx) per time column; record-break recovery is
// the rare divergent path; lane halves merged with shfl_xor(16).
//
// A (16x4 f32, 2 VGPRs): lanes 0-15 -> M=lane, regs {K=4q, 4q+1};
//                        lanes 16-31 -> M=lane-16, regs {K=4q+2, 4q+3}.
// B mirrors A over K rows; C/D: lane<16 -> M=r, lane>=16 -> M=r+8.
// ---------------------------------------------------------------------------
__global__ void __launch_bounds__(256) k_logits_argmax(
    const float* __restrict__ Wio, const float* __restrict__ bio,
    const float* __restrict__ Cmat, float* __restrict__ part) {
  const int lane = threadIdx.x & 31;
  const int wid  = threadIdx.x >> 5;      // vocab split 0..7
  const int half = (lane >= 16) ? 1 : 0;
  const int l16  = lane & 15;
  const int t0   = blockIdx.x * 32 + l16; // time tile pair: tt0, tt0+1
  const int t1   = t0 + 16;

  // B operands (Combined^T) for both time tiles: 5 K-chunks x 2 regs each.
  v2f b0[5], b1[5];
#pragma unroll
  for (int q = 0; q < 5; q++) {
    b0[q] = *(const v2f*)(Cmat + t0 * CW + 4 * q + 2 * half);
    b1[q] = *(const v2f*)(Cmat + t1 * CW + 4 * q + 2 * half);
  }

  float bestVal0 = -3.4e38f, bestVal1 = -3.4e38f;
  int   bestIdx0 = 0,        bestIdx1 = 0;

  int vBase = wid * TILES_PER_SPLIT * 16;
  const float* wp = Wio + (long)(vBase + l16) * CW + 2 * half;  // A rows
  const float* bb = bio + vBase + 8 * half;                     // bias

  for (int vt = 0; vt < TILES_PER_SPLIT; ++vt) {
    // shared per-vocab-tile operands: bias (2 x b128) + A (5 x b64)
    v4f c0 = *(const v4f*)(bb);
    v4f c1 = *(const v4f*)(bb + 4);
    v2f a[5];
#pragma unroll
    for (int q = 0; q < 5; q++) a[q] = *(const v2f*)(wp + 4 * q);

    v8f acc0, acc1;
    acc0[0] = c0[0]; acc0[1] = c0[1]; acc0[2] = c0[2]; acc0[3] = c0[3];
    acc0[4] = c1[0]; acc0[5] = c1[1]; acc0[6] = c1[2]; acc0[7] = c1[3];
    acc1 = acc0;

#pragma unroll
    for (int q = 0; q < 5; q++)
      acc0 = __builtin_amdgcn_wmma_f32_16x16x4_f32(
          false, a[q], false, b0[q], (short)0, acc0, false, false);
#pragma unroll
    for (int q = 0; q < 5; q++)
      acc1 = __builtin_amdgcn_wmma_f32_16x16x4_f32(
          false, a[q], false, b1[q], (short)0, acc1, false, false);

    // fast tile-max (folds to v_max3_num_f32), rare index-recovery path
    float m0 = fmaxf(fmaxf(fmaxf(acc0[0], acc0[1]), fmaxf(acc0[2], acc0[3])),
                     fmaxf(fmaxf(acc0[4], acc0[5]), fmaxf(acc0[6], acc0[7])));
    float m1 = fmaxf(fmaxf(fmaxf(acc1[0], acc1[1]), fmaxf(acc1[2], acc1[3])),
                     fmaxf(fmaxf(acc1[4], acc1[5]), fmaxf(acc1[6], acc1[7])));
    if (m0 > bestVal0) {
      bestVal0 = m0;
      int idx = 0;
#pragma unroll
      for (int r = 7; r >= 0; --r)        // descending: lowest index wins
        if (acc0[r] == m0) idx = vBase + 8 * half + r;
      bestIdx0 = idx;
    }
    if (m1 > bestVal1) {
      bestVal1 = m1;
      int idx = 0;
#pragma unroll
      for (int r = 7; r >= 0; --r)
        if (acc1[r] == m1) idx = vBase + 8 * half + r;
      bestIdx1 = idx;
    }

    wp += 16 * CW;
    bb += 16;
    vBase += 16;
  }

  // merge lane halves (same time column, vocab +0..7 vs +8..15)
  float oV0 = __shfl_xor(bestVal0, 16, 32);
  int   oI0 = __shfl_xor(bestIdx0, 16, 32);
  if (oV0 > bestVal0 || (oV0 == bestVal0 && oI0 < bestIdx0)) {
    bestVal0 = oV0; bestIdx0 = oI0;
  }
  float oV1 = __shfl_xor(bestVal1, 16, 32);
  int   oI1 = __shfl_xor(bestIdx1, 16, 32);
  if (oV1 > bestVal1 || (oV1 == bestVal1 && oI1 < bestIdx1)) {
    bestVal1 = oV1; bestIdx1 = oI1;
  }
  if (lane < 16) {
    part[(t0 * NSPLIT + wid) * 2 + 0] = bestVal0;
    part[(t0 * NSPLIT + wid) * 2 + 1] = (float)bestIdx0;  // < 2^24, exact
    part[(t1 * NSPLIT + wid) * 2 + 0] = bestVal1;
    part[(t1 * NSPLIT + wid) * 2 + 1] = (float)bestIdx1;
  }
}

// ---------------------------------------------------------------------------
// Kernel 4: reduce the 8 vocab-split partials per timestep (splits are in
// ascending vocab order; tie -> lower index, matching jnp.argmax).
// ---------------------------------------------------------------------------
__global__ void k_reduce(const float* __restrict__ part,
                         float* __restrict__ out) {
  int t = blockIdx.x * blockDim.x + threadIdx.x;
  if (t >= SEQ_LEN) return;
  float bv = -3.4e38f, bi = 0.0f;
#pragma unroll
  for (int s = 0; s < NSPLIT; s++) {
    float v = part[(t * NSPLIT + s) * 2 + 0];
    float i = part[(t * NSPLIT + s) * 2 + 1];
    if (v > bv || (v == bv && i < bi)) { bv = v; bi = i; }
  }
  out[t] = bi;
}

// ---------------------------------------------------------------------------
extern "C" void kernel_launch(void* const* d_in, const int* in_sizes, int n_in,
                              void* d_out, int out_size, void* d_ws,
                              size_t ws_size, hipStream_t stream) {
  const int*   words = (const int*)d_in[0];    // (8192,1) int32
  const float* emb   = (const float*)d_in[1];  // (128000,10)
  const float* W_i2h = (const float*)d_in[2];  // (10,20)
  const float* b_i2h = (const float*)d_in[3];  // (10,)
  const float* W_i2o = (const float*)d_in[4];  // (128000,20)
  const float* b_i2o = (const float*)d_in[5];  // (128000,)
  float* out = (float*)d_out;                  // (8192,)

  float* Cmat = (float*)d_ws;                  // 8192*20 f32 (combined)
  float* cvec = Cmat + SEQ_LEN * CW;           // 8192*10 f32
  float* part = cvec + SEQ_LEN * HID;          // 8192*8*2 f32 partial argmax

  k_embed<<<SEQ_LEN / 256, 256, 0, stream>>>(words, emb, W_i2h, b_i2h, Cmat,
                                             cvec);
  k_recur<<<1, 32, 0, stream>>>(W_i2h, cvec, Cmat);
  k_logits_argmax<<<TTILES / 2, 256, 0, stream>>>(W_i2o, b_i2o, Cmat, part);
  k_reduce<<<SEQ_LEN / 256, 256, 0, stream>>>(part, out);
}